// RelScores_68375879352672
// MI455X (gfx1250) — compile-verified
//
#include <hip/hip_runtime.h>
#include <hip/hip_bf16.h>

typedef __attribute__((ext_vector_type(2))) float v2f;
typedef __attribute__((ext_vector_type(8))) float v8f;

#define S_ 16
#define N_ 64
#define D_ 1270
#define F_ 150
#define KTILE 64
#define NKT 20            // ceil(1270/64)
#define DPAD 1280         // NKT*64
#define OUT_ELEMS (16*64*64*8)

// ---------------- Stage 1a: M2 = W_h @ W_out  (150x7, padded to 150x8) ----
__global__ void k_prep1(const float* __restrict__ Wh, const float* __restrict__ Wout,
                        float* __restrict__ M2, float* __restrict__ loss_slot) {
    if (blockIdx.x == 0 && threadIdx.x == 0) *loss_slot = 0.0f;
    for (int id = threadIdx.x; id < F_ * 8; id += 256) {
        int f = id >> 3, k = id & 7;
        float acc = 0.0f;
        if (k < 7) {
            const float* whf = Wh + (size_t)f * F_;
            for (int g = 0; g < F_; ++g) acc += whf[g] * Wout[g * 7 + k];
        }
        M2[id] = acc;
    }
}

// ---------------- Stage 1b: G^T (16 x 1280, zero padded), A1, A2, c2 ------
__global__ void k_prep2(const float* __restrict__ Win, const float* __restrict__ bin,
                        const float* __restrict__ bh, const float* __restrict__ bout,
                        const float* __restrict__ Wout, const float* __restrict__ M2,
                        float* __restrict__ Gpt, float* __restrict__ A1p,
                        float* __restrict__ A2p, float* __restrict__ c2) {
    const int NG = 16 * DPAD;       // 20480
    const int NA = DPAD * 8;        // 10240
    int id = blockIdx.x * 256 + threadIdx.x;
    if (id < NG) {
        int n = id / DPAD, dk = id % DPAD;
        float acc = 0.0f;
        if (n < 7 && dk < D_) {
            const float* w3 = Win + (size_t)(2 * D_ + dk) * F_;
            for (int f = 0; f < F_; ++f) acc += w3[f] * M2[f * 8 + n];
        }
        Gpt[n * DPAD + dk] = acc;   // transposed, zero-padded: no bounds checks later
    } else if (id < NG + 2 * NA) {
        int id2 = id - NG;
        int which = id2 / NA, t = id2 % NA;
        int d = t >> 3, k = t & 7;
        float acc = 0.0f;
        if (d < D_ && k < 7) {
            const float* w = Win + (size_t)(which * D_ + d) * F_;
            for (int f = 0; f < F_; ++f) acc += w[f] * M2[f * 8 + k];
        }
        (which ? A2p : A1p)[t] = acc;
    } else if (id < NG + 2 * NA + 8) {
        int k = id - (NG + 2 * NA);
        float acc = 0.0f;
        if (k < 7) {
            for (int f = 0; f < F_; ++f) acc += bin[f] * M2[f * 8 + k];
            for (int g = 0; g < F_; ++g) acc += bh[g] * Wout[g * 7 + k];
            acc += bout[k];
        }
        c2[k] = acc;
    }
}

// ---------------- Stage 2: per-entity terms a = e@A1, b = e@A2 ------------
__global__ void k_rowproj(const float* __restrict__ emb, const float* __restrict__ A1p,
                          const float* __restrict__ A2p, float* __restrict__ a_buf,
                          float* __restrict__ b_buf) {
    int id = blockIdx.x * 256 + threadIdx.x;   // 1024 rows * 8 k-slots
    if (id >= S_ * N_ * 8) return;
    int row = id >> 3, k = id & 7;
    const float* e = emb + (size_t)row * D_;
    float sa = 0.0f, sb = 0.0f;
    for (int d = 0; d < D_; ++d) {
        float ev = e[d];
        sa += ev * A1p[d * 8 + k];   // k==7 slots are zero-filled -> sa stays 0
        sb += ev * A2p[d * 8 + k];
    }
    a_buf[id] = sa;
    b_buf[id] = sb;
}

// ---------------- Stage 3 (hot): per-(s,j) pair GEMM with f32 WMMA --------
// scores[s, 0:64, j, 0:7] = (E_s * e_j) @ G  (+ row/col/bias terms)
// One workgroup per (s,j). 4 waves, each owns one 16x16 M-tile of the 64x16
// output. K iterated in 64-wide LDS-staged tiles; e_j folded in at staging.
__global__ __launch_bounds__(128) void k_pair(
    const float* __restrict__ emb, const float* __restrict__ escore,
    const float* __restrict__ Gpt, const float* __restrict__ a_buf,
    const float* __restrict__ b_buf, const float* __restrict__ c2,
    float* __restrict__ out) {
    __shared__ __align__(16) float Es[64 * 68];   // E_s(i, k) * e_j(k), padded stride
    __shared__ __align__(16) float Gt[16 * 68];   // G^T(n, k)

    const int t = threadIdx.x;
    const int sj = blockIdx.x;          // s*64 + j
    const int s = sj >> 6, j = sj & 63;
    const float* Eb = emb + (size_t)s * N_ * D_;
    const float* ejp = Eb + (size_t)j * D_;

    const int lane = t & 31;
    const int wave = t >> 5;
    const int half = lane >> 4;         // 0: K lo pair / M 0..7 ; 1: K hi pair / M 8..15
    const int ln = lane & 15;           // A: row m ; B/C: col n
    const int m0 = wave * 16;

    v8f acc = {0.f, 0.f, 0.f, 0.f, 0.f, 0.f, 0.f, 0.f};

    for (int kt = 0; kt < NKT; ++kt) {
        const int kb = kt * KTILE;
        // Stage A-tile: Es[r][c] = E_s[r][kb+c] * e_j[kb+c]  (zero past D)
        for (int q = 0; q < 32; ++q) {
            int idx = t + q * 128;
            int r = idx >> 6, c = idx & 63;
            int d = kb + c;
            float v = 0.0f;
            if (d < D_) v = Eb[(size_t)r * D_ + d] * ejp[d];
            Es[r * 68 + c] = v;
        }
        // Stage B-tile: Gt[n][c] = G^T[n][kb+c]  (already zero-padded to 1280)
        for (int q = 0; q < 8; ++q) {
            int idx = t + q * 128;
            int r = idx >> 6, c = idx & 63;
            Gt[r * 68 + c] = Gpt[r * DPAD + kb + c];
        }
        __syncthreads();

        // 16 WMMA k-steps of 4 each. A[m][k]: lane = m + 16*(k>>1), vgpr = k&1.
        // B[k][n]: lane = n + 16*(k>>1), vgpr = k&1.
#pragma unroll
        for (int kk = 0; kk < 16; ++kk) {
            const int col = kk * 4 + half * 2;
            const v2f A = *reinterpret_cast<const v2f*>(&Es[(m0 + ln) * 68 + col]);
            const v2f B = *reinterpret_cast<const v2f*>(&Gt[ln * 68 + col]);
            acc = __builtin_amdgcn_wmma_f32_16x16x4_f32(
                false, A, false, B, (short)0, acc, false, false);
        }
        __syncthreads();
    }

    // Epilogue. C layout: vgpr r -> M = m0 + 8*half + r, N = ln.
    const float esj = escore[sj];
    const float bn = (ln < 7) ? (b_buf[sj * 8 + ln] + c2[ln] + esj) : 0.0f;
#pragma unroll
    for (int r = 0; r < 8; ++r) {
        const int i = m0 + half * 8 + r;
        const int row = s * N_ + i;
        const size_t base = (((size_t)row) * N_ + j) * 8;
        if (ln < 7) {
            float v = acc[r] + a_buf[row * 8 + ln] + escore[row] + bn;
            out[base + 1 + ln] = v;
        } else if (ln == 7) {
            out[base] = 0.0f;        // dummy class-0 channel
        }
    }
}

// ---------------- Stage 4: masked CE loss over rel_scores -----------------
__global__ void k_loss(const float* __restrict__ scores, const int* __restrict__ labels,
                       const int* __restrict__ npe, float* __restrict__ loss) {
    __shared__ float red[256];
    int id = blockIdx.x * 256 + threadIdx.x;   // 65536 = S*N*N
    float ce = 0.0f;
    {
        int s = id >> 12;
        int i = (id >> 6) & 63;
        int j = id & 63;
        const float* x = scores + (size_t)id * 8;
        float m = x[0];
#pragma unroll
        for (int k = 1; k < 8; ++k) m = fmaxf(m, x[k]);
        float sum = 0.0f;
#pragma unroll
        for (int k = 0; k < 8; ++k) sum += expf(x[k] - m);
        float lse = m + logf(sum);
        int lab = labels[id] & 7;
        float cev = lse - x[lab];
        if (i < npe[s] && j < npe[s]) ce = cev;
    }
    red[threadIdx.x] = ce;
    __syncthreads();
    for (int off = 128; off > 0; off >>= 1) {
        if (threadIdx.x < off) red[threadIdx.x] += red[threadIdx.x + off];
        __syncthreads();
    }
    if (threadIdx.x == 0) atomicAdd(loss, red[0]);
}

extern "C" void kernel_launch(void* const* d_in, const int* in_sizes, int n_in,
                              void* d_out, int out_size, void* d_ws, size_t ws_size,
                              hipStream_t stream) {
    const float* emb    = (const float*)d_in[0];   // (16,64,1270)
    const float* escore = (const float*)d_in[1];   // (16,64)
    const float* Win    = (const float*)d_in[2];   // (3810,150)
    const float* bin    = (const float*)d_in[3];   // (150,)
    const float* Wh     = (const float*)d_in[4];   // (150,150)
    const float* bh     = (const float*)d_in[5];   // (150,)
    const float* Wout   = (const float*)d_in[6];   // (150,7)
    const float* bout   = (const float*)d_in[7];   // (7,)
    const int*   labels = (const int*)d_in[8];     // (16,64,64)
    const int*   npe    = (const int*)d_in[9];     // (16,)
    float* out = (float*)d_out;                    // 524288 rel_scores + 1 loss

    float* ws    = (float*)d_ws;
    float* M2    = ws;                 // 1280 (150x8 used)
    float* Gpt   = M2 + 1280;          // 16 x 1280
    float* A1p   = Gpt + 16 * DPAD;    // 1280 x 8
    float* A2p   = A1p + DPAD * 8;     // 1280 x 8
    float* a_buf = A2p + DPAD * 8;     // 1024 x 8
    float* b_buf = a_buf + S_ * N_ * 8;
    float* c2    = b_buf + S_ * N_ * 8; // 8
    float* loss  = out + OUT_ELEMS;

    k_prep1<<<1, 256, 0, stream>>>(Wh, Wout, M2, loss);
    {
        int total = 16 * DPAD + 2 * DPAD * 8 + 8;
        k_prep2<<<(total + 255) / 256, 256, 0, stream>>>(Win, bin, bh, bout, Wout, M2,
                                                         Gpt, A1p, A2p, c2);
    }
    k_rowproj<<<(S_ * N_ * 8 + 255) / 256, 256, 0, stream>>>(emb, A1p, A2p, a_buf, b_buf);
    k_pair<<<S_ * N_, 128, 0, stream>>>(emb, escore, Gpt, a_buf, b_buf, c2, out);
    k_loss<<<S_ * N_ * N_ / 256, 256, 0, stream>>>(out, labels, npe, loss);
}